// NTMWriteHead_28827820491518
// MI455X (gfx1250) — compile-verified
//
#include <hip/hip_runtime.h>
#include <hip/hip_bf16.h>
#include <math.h>

// Problem constants (match the reference)
#define BDIM   1024
#define NDIM   512
#define MDIM   128
#define CDIM_  1024
#define OUT_DIM (3 * MDIM + 6)   // 390

typedef __attribute__((ext_vector_type(2))) float v2f;
typedef __attribute__((ext_vector_type(4))) float v4f;
typedef __attribute__((ext_vector_type(8))) float v8f;

// ---------------------------------------------------------------------------
// Kernel 1: o = hid @ W_fc^T + b_fc   (fp32 GEMM via V_WMMA_F32_16X16X4_F32)
//   hid  : [B, CDIM]       row-major
//   W_fc : [OUT_DIM, CDIM] row-major  -> B matrix element (k, n) = W_fc[n*CDIM + k]
//   o    : [B, OUT_DIM]
// One wave computes one 16x16 tile of o. K-loop steps of 4 (f32 WMMA shape).
// WMMA VGPR layouts (CDNA5 ISA 7.12.2):
//   A 16x4 f32 : lanes 0-15 -> {K=0,K=1}, lanes 16-31 -> {K=2,K=3}, M = lane%16
//   B  4x16 f32: mirrored,  N = lane%16
//   C/D 16x16  : col = lane%16, row = vgpr + (lane>=16 ? 8 : 0)
// Operands stay L2-resident (5.6 MB total) with default RT cache policy.
// ---------------------------------------------------------------------------
__global__ __launch_bounds__(128)
void fc_wmma_kernel(const float* __restrict__ hid,
                    const float* __restrict__ Wfc,
                    const float* __restrict__ bfc,
                    float* __restrict__ o) {
  const int lane = threadIdx.x & 31;
  const int wave = threadIdx.x >> 5;
  const int m0 = blockIdx.x * 16;                  // batch-row tile base
  const int n0 = (blockIdx.y * 4 + wave) * 16;     // out-col tile base
  if (n0 >= OUT_DIM) return;                       // wave-uniform exit: EXEC stays full per-wave

  const int half = lane >> 4;                      // 0: lanes 0-15, 1: lanes 16-31
  const int l16  = lane & 15;
  const int arow = m0 + l16;                       // A row owned by this lane
  const int bcol = n0 + l16;                       // B col owned by this lane
  const int bcolc = (bcol < OUT_DIM) ? bcol : (OUT_DIM - 1);  // clamp loads in partial tile

  const float* __restrict__ arp = hid + (size_t)arow  * CDIM_;
  const float* __restrict__ brp = Wfc + (size_t)bcolc * CDIM_;

  v8f acc = {};
  for (int k0 = 0; k0 < CDIM_; k0 += 4) {
    const int ka = k0 + half * 2;
    v2f a; a.x = arp[ka]; a.y = arp[ka + 1];
    v2f b; b.x = brp[ka]; b.y = brp[ka + 1];
    acc = __builtin_amdgcn_wmma_f32_16x16x4_f32(
        /*neg_a=*/false, a, /*neg_b=*/false, b,
        /*c_mod=*/(short)0, acc, /*reuse_a=*/false, /*reuse_b=*/false);
  }

  if (bcol < OUT_DIM) {
    const float bias = bfc[bcol];
#pragma unroll
    for (int r = 0; r < 8; ++r) {
      const int row = m0 + r + half * 8;
      o[(size_t)row * OUT_DIM + bcol] = acc[r] + bias;
    }
  }
}

// ---------------------------------------------------------------------------
// Reductions (wave32; warpSize == 32 on gfx1250)
// ---------------------------------------------------------------------------
__device__ __forceinline__ float waveReduceSum(float v) {
#pragma unroll
  for (int off = 16; off > 0; off >>= 1) v += __shfl_xor(v, off, 32);
  return v;
}
__device__ __forceinline__ float waveReduceMax(float v) {
#pragma unroll
  for (int off = 16; off > 0; off >>= 1) v = fmaxf(v, __shfl_xor(v, off, 32));
  return v;
}
__device__ __forceinline__ float blockReduceSum(float v, float* red, int tid) {
  const int lane = tid & 31, wid = tid >> 5;
  v = waveReduceSum(v);
  if (lane == 0) red[wid] = v;
  __syncthreads();
  if (wid == 0) {
    float r = (lane < 8) ? red[lane] : 0.0f;
    r = waveReduceSum(r);
    if (lane == 0) red[0] = r;
  }
  __syncthreads();
  const float out = red[0];
  __syncthreads();
  return out;
}
__device__ __forceinline__ float blockReduceMax(float v, float* red, int tid) {
  const int lane = tid & 31, wid = tid >> 5;
  v = waveReduceMax(v);
  if (lane == 0) red[wid] = v;
  __syncthreads();
  if (wid == 0) {
    float r = (lane < 8) ? red[lane] : -3.402823466e+38f;
    r = waveReduceMax(r);
    if (lane == 0) red[0] = r;
  }
  __syncthreads();
  const float out = red[0];
  __syncthreads();
  return out;
}
__device__ __forceinline__ float softplusf(float x) {
  // numerically stable softplus
  return (x > 0.0f) ? (x + log1pf(__expf(-x))) : log1pf(__expf(x));
}
__device__ __forceinline__ float sigmoidf_(float x) {
  return 1.0f / (1.0f + __expf(-x));
}

// ---------------------------------------------------------------------------
// Kernel 2: fused content addressing + location addressing + erase/add write.
// One block (256 threads = 8 waves) per batch element b.
// memory[b] (512x128 f32 = 256 KB) is cached in dynamic LDS on the first pass
// (CDNA5 WGP LDS is 320 KB) so the erase/add pass never re-reads HBM.
// The memory stream (read once, write once, zero cross-block reuse) uses
// non-temporal cache policy (TH=NT) so 512 MB of streaming traffic does not
// churn the 192 MB L2 out from under the L2-resident GEMM operands.
//
// Dynamic LDS layout (floats):
//   mem_s  [NDIM*MDIM] = 65536
//   k_s    [MDIM], e_s [MDIM], a_s [MDIM]
//   wc_s   [NDIM], wg_s [NDIM]
//   red_s  [16], sc_s [8]
// ---------------------------------------------------------------------------
__global__ __launch_bounds__(256)
void ntm_head_kernel(const float* __restrict__ o,
                     const float* __restrict__ w_pre,
                     const float* __restrict__ memory,
                     float* __restrict__ w_out,
                     float* __restrict__ mem_out) {
  extern __shared__ float smem[];
  float* mem_s = smem;                         // 65536
  float* k_s   = mem_s + NDIM * MDIM;          // 128
  float* e_s   = k_s + MDIM;                   // 128
  float* a_s   = e_s + MDIM;                   // 128
  float* wc_s  = a_s + MDIM;                   // 512
  float* wg_s  = wc_s + NDIM;                  // 512
  float* red_s = wg_s + NDIM;                  // 16
  float* sc_s  = red_s + 16;                   // 8: beta,g,s0,s1,s2,gamma,knorm

  const int b    = blockIdx.x;
  const int tid  = threadIdx.x;
  const int lane = tid & 31;
  const int wid  = tid >> 5;                   // 0..7
  const float eps = 1e-16f;

  const float* __restrict__ ob = o + (size_t)b * OUT_DIM;

  // ---- head params + activations ----
  if (tid < MDIM) {
    k_s[tid] = ob[tid] + eps;                          // k + eps
    e_s[tid] = sigmoidf_(ob[MDIM + 6 + tid]);          // erase vector
    a_s[tid] = ob[2 * MDIM + 6 + tid];                 // add vector
  }
  if (tid == 0) {
    sc_s[0] = softplusf(ob[MDIM]);                     // beta
    sc_s[1] = sigmoidf_(ob[MDIM + 1]);                 // g
    const float s0r = ob[MDIM + 2], s1r = ob[MDIM + 3], s2r = ob[MDIM + 4];
    const float sm = fmaxf(s0r, fmaxf(s1r, s2r));
    const float es0 = __expf(s0r - sm), es1 = __expf(s1r - sm), es2 = __expf(s2r - sm);
    const float inv = 1.0f / (es0 + es1 + es2);
    sc_s[2] = es0 * inv; sc_s[3] = es1 * inv; sc_s[4] = es2 * inv;
    sc_s[5] = 1.0f + softplusf(ob[MDIM + 5]);          // gamma
  }
  __syncthreads();

  // ---- ||k + eps|| (wave 0) ----
  if (wid == 0) {
    const float c0 = k_s[lane], c1 = k_s[lane + 32], c2 = k_s[lane + 64], c3 = k_s[lane + 96];
    float sq = c0 * c0 + c1 * c1 + c2 * c2 + c3 * c3;
    sq = waveReduceSum(sq);
    if (lane == 0) sc_s[6] = sqrtf(sq);
  }
  __syncthreads();

  const float beta  = sc_s[0];
  const float knorm = sc_s[6];

  // ---- content pass: stream memory[b] once (NT), cache in LDS, fuse dot+norm ----
  const float* __restrict__ memb = memory + (size_t)b * NDIM * MDIM;
  const v4f kv = ((const v4f*)k_s)[lane];
  for (int n = wid; n < NDIM; n += 8) {
    const v4f mv =
        __builtin_nontemporal_load((const v4f*)(memb + (size_t)n * MDIM) + lane);
    ((v4f*)(mem_s + n * MDIM))[lane] = mv;             // cache raw memory row in LDS
    const float mx = mv.x + eps, my = mv.y + eps, mz = mv.z + eps, mw = mv.w + eps;
    float num = mx * kv.x + my * kv.y + mz * kv.z + mw * kv.w;
    float sq  = mx * mx + my * my + mz * mz + mw * mw;
    num = waveReduceSum(num);
    sq  = waveReduceSum(sq);
    if (lane == 0) {
      const float denom = sqrtf(sq) * knorm;
      wc_s[n] = beta * (num / fmaxf(denom, 1e-8f));    // beta * K (pre-softmax logits)
    }
  }
  __syncthreads();

  // ---- softmax over N=512 (each thread owns 2 rows) ----
  const int i0 = tid, i1 = tid + 256;
  const float l0 = wc_s[i0], l1 = wc_s[i1];
  const float mmax = blockReduceMax(fmaxf(l0, l1), red_s, tid);
  const float x0 = __expf(l0 - mmax), x1 = __expf(l1 - mmax);
  const float xsum = blockReduceSum(x0 + x1, red_s, tid);
  const float xinv = 1.0f / xsum;

  // ---- interpolate with w_pre ----
  const float g = sc_s[1];
  const float* __restrict__ wpb = w_pre + (size_t)b * NDIM;
  wg_s[i0] = g * (x0 * xinv) + (1.0f - g) * wpb[i0];
  wg_s[i1] = g * (x1 * xinv) + (1.0f - g) * wpb[i1];
  __syncthreads();

  // ---- circular shift + sharpen + renormalize ----
  const float s0 = sc_s[2], s1 = sc_s[3], s2 = sc_s[4], gamma = sc_s[5];
  const float sh0 = s0 * wg_s[(i0 + NDIM - 1) & (NDIM - 1)] + s1 * wg_s[i0]
                  + s2 * wg_s[(i0 + 1) & (NDIM - 1)];
  const float sh1 = s0 * wg_s[(i1 + NDIM - 1) & (NDIM - 1)] + s1 * wg_s[i1]
                  + s2 * wg_s[(i1 + 1) & (NDIM - 1)];
  const float p0 = __powf(sh0, gamma);
  const float p1 = __powf(sh1, gamma);
  const float psum = blockReduceSum(p0 + p1, red_s, tid);
  const float pinv = 1.0f / (psum + eps);
  const float w0 = p0 * pinv, w1 = p1 * pinv;
  w_out[(size_t)b * NDIM + i0] = w0;
  w_out[(size_t)b * NDIM + i1] = w1;
  wc_s[i0] = w0;                                       // reuse wc_s to hold final w
  wc_s[i1] = w1;
  __syncthreads();

  // ---- erase/add write: memory rows come from LDS; stores stream NT to HBM ----
  const v4f ev = ((const v4f*)e_s)[lane];
  const v4f av = ((const v4f*)a_s)[lane];
  float* __restrict__ mob = mem_out + (size_t)b * NDIM * MDIM;
  for (int n = wid; n < NDIM; n += 8) {
    const float wn = wc_s[n];
    const v4f mv = ((const v4f*)(mem_s + n * MDIM))[lane];
    v4f r;
    r.x = mv.x * (1.0f - wn * ev.x) + wn * av.x;
    r.y = mv.y * (1.0f - wn * ev.y) + wn * av.y;
    r.z = mv.z * (1.0f - wn * ev.z) + wn * av.z;
    r.w = mv.w * (1.0f - wn * ev.w) + wn * av.w;
    __builtin_nontemporal_store(r, (v4f*)(mob + (size_t)n * MDIM) + lane);
  }
}

// ---------------------------------------------------------------------------
// Launcher
// inputs: hid[B,CDIM], w_pre[B,N], memory[B,N,M], W_fc[OUT_DIM,CDIM], b_fc[OUT_DIM]
// output: w[B,N] then mem_new[B,N,M], concatenated flat in d_out.
// d_ws: o[B,OUT_DIM] scratch (1.6 MB).
// ---------------------------------------------------------------------------
extern "C" void kernel_launch(void* const* d_in, const int* in_sizes, int n_in,
                              void* d_out, int out_size, void* d_ws, size_t ws_size,
                              hipStream_t stream) {
  const float* hid    = (const float*)d_in[0];
  const float* w_pre  = (const float*)d_in[1];
  const float* memory = (const float*)d_in[2];
  const float* W_fc   = (const float*)d_in[3];
  const float* b_fc   = (const float*)d_in[4];

  float* o = (float*)d_ws;                      // [B, OUT_DIM]
  float* w_out   = (float*)d_out;               // [B, N]
  float* mem_out = w_out + (size_t)BDIM * NDIM; // [B, N, M]

  // GEMM: 64 row tiles x ceil(25/4)=7 col groups, 4 waves (col tiles) per block
  dim3 g1(BDIM / 16, (OUT_DIM + 63) / 64);
  fc_wmma_kernel<<<g1, 128, 0, stream>>>(hid, W_fc, b_fc, o);

  // Fused head: one block per batch element, 256 KB + small LDS
  const size_t lds_bytes =
      (size_t)(NDIM * MDIM + 3 * MDIM + 2 * NDIM + 16 + 8) * sizeof(float);
  ntm_head_kernel<<<BDIM, 256, lds_bytes, stream>>>(o, w_pre, memory, w_out, mem_out);
}